// ListMLE_10531259809808
// MI455X (gfx1250) — compile-verified
//
#include <hip/hip_runtime.h>

// Problem constants (match reference).
constexpr int B_ROWS  = 8192;
constexpr int N_COLS  = 4096;
constexpr int TPB     = 512;            // 16 wave32 waves per block
constexpr int KPT     = N_COLS / TPB;   // 8 elements per thread
constexpr int NWAVES  = TPB / 32;       // 16

// -------- CDNA5 async global->LDS copy (b128 per lane) --------------------
typedef int v4i_t __attribute__((ext_vector_type(4)));

__device__ __forceinline__ void async_load_b128(const float* g, unsigned lds_off) {
#if __has_builtin(__builtin_amdgcn_global_load_async_to_lds_b128)
    typedef __attribute__((address_space(1))) v4i_t gvec_t;
    typedef __attribute__((address_space(3))) v4i_t lvec_t;
    __builtin_amdgcn_global_load_async_to_lds_b128(
        (gvec_t*)(unsigned long long)g, (lvec_t*)lds_off, 0, 0);
#else
    // vdst = 32-bit LDS byte address, vaddr = 64-bit global address
    asm volatile("global_load_async_to_lds_b128 %0, %1, off"
                 :: "v"(lds_off), "v"(g) : "memory");
#endif
}

__device__ __forceinline__ void wait_async0() {
#if __has_builtin(__builtin_amdgcn_s_wait_asynccnt)
    __builtin_amdgcn_s_wait_asynccnt(0);
#else
    asm volatile("s_wait_asynccnt 0" ::: "memory");
#endif
}

// -------- logsumexp scan state: (m, s); identity encoded as s == 0 --------
__device__ __forceinline__ void lse_merge(float& m, float& s, float mo, float so) {
    if (so == 0.0f) return;
    if (s == 0.0f) { m = mo; s = so; return; }
    if (mo > m) { s = s * __expf(m - mo) + so; m = mo; }
    else        { s = s + so * __expf(mo - m); }
}

__global__ __launch_bounds__(TPB)
void listmle_row_kernel(const float* __restrict__ outputs,
                        const int*   __restrict__ labels,
                        float*       __restrict__ partials) {
    __shared__ float row[N_COLS];
    __shared__ float wm[NWAVES], ws[NWAVES];
    __shared__ float red[NWAVES];

    const int b    = blockIdx.x;
    const int t    = threadIdx.x;
    const int lane = t & 31;
    const int wid  = t >> 5;

    // ---- stage the row into LDS with async copies ----
    const float* grow = outputs + (size_t)b * N_COLS;
    #pragma unroll
    for (int p = 0; p < N_COLS / (TPB * 4); ++p) {   // 2 passes
        const int idx = (p * TPB + t) * 4;
        async_load_b128(grow + idx, (unsigned)(unsigned long long)&row[idx]);
    }
    wait_async0();
    __syncthreads();

    // ---- gather + in-thread sequential scan over 8 elements ----
    const int* lrow = labels + (size_t)b * N_COLS + t * KPT;
    const int4 l0 = ((const int4*)lrow)[0];
    const int4 l1 = ((const int4*)lrow)[1];
    const int idxs[KPT] = { l0.x, l0.y, l0.z, l0.w, l1.x, l1.y, l1.z, l1.w };

    float xm[KPT], xs[KPT];
    float rm = 0.0f, rs = 0.0f;   // identity
    #pragma unroll
    for (int k = 0; k < KPT; ++k) {
        const float x = row[idxs[k] & (N_COLS - 1)];
        if (rs == 0.0f)      { rm = x; rs = 1.0f; }
        else if (x > rm)     { rs = rs * __expf(rm - x) + 1.0f; rm = x; }
        else                 { rs += __expf(x - rm); }
        xm[k] = rm; xs[k] = rs;
    }

    // un-permuted outputs sum (contiguous chunk of the LDS row)
    float osum = 0.0f;
    #pragma unroll
    for (int k = 0; k < KPT; ++k) osum += row[t * KPT + k];

    // ---- wave32 inclusive shuffle scan of thread aggregates ----
    float sm = rm, ss = rs;
    #pragma unroll
    for (int d = 1; d < 32; d <<= 1) {
        const float mo = __shfl_up(sm, d, 32);
        const float so = __shfl_up(ss, d, 32);
        if (lane >= d) lse_merge(sm, ss, mo, so);
    }
    if (lane == 31) { wm[wid] = sm; ws[wid] = ss; }
    __syncthreads();

    // ---- cross-wave scan (16 aggregates) done by wave 0 ----
    if (wid == 0) {
        float am = (lane < NWAVES) ? wm[lane] : 0.0f;
        float as = (lane < NWAVES) ? ws[lane] : 0.0f;
        #pragma unroll
        for (int d = 1; d < NWAVES; d <<= 1) {
            const float mo = __shfl_up(am, d, 32);
            const float so = __shfl_up(as, d, 32);
            if (lane >= d && lane < NWAVES) lse_merge(am, as, mo, so);
        }
        if (lane < NWAVES) { wm[lane] = am; ws[lane] = as; }
    }
    __syncthreads();

    // ---- exclusive prefix = (wave-exclusive) ⊕ (thread-exclusive in wave) ----
    float em = 0.0f, es = 0.0f;
    if (wid > 0) { em = wm[wid - 1]; es = ws[wid - 1]; }
    float tm = __shfl_up(sm, 1, 32);
    float tv = __shfl_up(ss, 1, 32);
    if (lane == 0) { tm = 0.0f; tv = 0.0f; }
    lse_merge(em, es, tm, tv);

    // ---- apply prefix, score = m + log(s), accumulate ----
    float ssum = 0.0f;
    #pragma unroll
    for (int k = 0; k < KPT; ++k) {
        float M = xm[k], S = xs[k];
        if (es != 0.0f) {
            if (em > M) { S = S * __expf(M - em) + es; M = em; }
            else        { S = S + es * __expf(em - M); }
        }
        ssum += M + __logf(S);
    }

    float part = ssum - osum;
    #pragma unroll
    for (int d = 16; d > 0; d >>= 1) part += __shfl_down(part, d, 32);
    if (lane == 0) red[wid] = part;
    __syncthreads();
    if (t == 0) {
        float tot = 0.0f;
        #pragma unroll
        for (int i = 0; i < NWAVES; ++i) tot += red[i];
        partials[b] = tot;
    }
}

__global__ __launch_bounds__(512)
void listmle_reduce_kernel(const float* __restrict__ partials,
                           float*       __restrict__ out) {
    __shared__ double red[16];
    const int t = threadIdx.x, lane = t & 31, wid = t >> 5;
    double acc = 0.0;
    for (int i = t; i < B_ROWS; i += 512) acc += (double)partials[i];
    #pragma unroll
    for (int d = 16; d > 0; d >>= 1) acc += __shfl_down(acc, d, 32);
    if (lane == 0) red[wid] = acc;
    __syncthreads();
    if (t == 0) {
        double tot = 0.0;
        #pragma unroll
        for (int i = 0; i < 16; ++i) tot += red[i];
        out[0] = (float)(tot / ((double)B_ROWS * (double)N_COLS));
    }
}

extern "C" void kernel_launch(void* const* d_in, const int* in_sizes, int n_in,
                              void* d_out, int out_size, void* d_ws, size_t ws_size,
                              hipStream_t stream) {
    (void)in_sizes; (void)n_in; (void)out_size; (void)ws_size;
    const float* outputs = (const float*)d_in[0];
    const int*   labels  = (const int*)d_in[1];   // JAX default x64-off => int32
    float* partials = (float*)d_ws;               // 8192 floats (32 KB) scratch
    float* out      = (float*)d_out;

    listmle_row_kernel<<<B_ROWS, TPB, 0, stream>>>(outputs, labels, partials);
    listmle_reduce_kernel<<<1, 512, 0, stream>>>(partials, out);
}